// ObjectCondensationLoss_10857677324663
// MI455X (gfx1250) — compile-verified
//
#include <hip/hip_runtime.h>
#include <stdint.h>
#include <math.h>

// ---------------------------------------------------------------------------
// ObjectCondensationLoss for MI455X (gfx1250, wave32)
//  - N x K distance pass uses V_WMMA_F32_16X16X4_F32 (2 WMMAs cover D=8)
//  - per-object argmax via packed u64 atomicMax (first-index tie-break)
//  - wave32 butterfly reductions via ds_swizzle_b32 (group-of-32 XOR mode)
//  - WGP-scope global_prefetch_b8 for the next row tile
// ---------------------------------------------------------------------------

#define N_HITS 120000
#define K_OBJ  512
#define D_DIM  8
#define C_CLS  6
#define EPSF   1e-9f

typedef float v2f __attribute__((ext_vector_type(2)));
typedef float v8f __attribute__((ext_vector_type(8)));

// ---- workspace layout (bytes) ----
#define OFF_ACC     0                              // 16 u32 accumulator slots
#define OFF_KEYS    64                             // K u64 argmax keys
#define OFF_ATTCNT  (OFF_KEYS + K_OBJ*8)           // K u32 hits-per-object
#define OFF_QK      (OFF_ATTCNT + K_OBJ*4)         // K f32 q at condensation pt
#define OFF_INVATT  (OFF_QK + K_OBJ*4)             // K f32 1/(att_norm+eps)
#define OFF_INVNREP (OFF_INVATT + K_OBJ*4)         // K f32 1/(n_rep_k+eps)
#define OFF_XKN2    (OFF_INVNREP + K_OBJ*4)        // K f32 ||x_k||^2
#define OFF_XK      (OFF_XKN2 + K_OBJ*4)           // K*8 f32 condensation coords
#define OFF_Q       (OFF_XK + K_OBJ*D_DIM*4)       // N f32 q per hit
#define OFF_XN2     (OFF_Q + N_HITS*4)             // N f32 ||x_i||^2
// total = 990784 bytes

// accumulator slot indices (u32/f32 view)
//  0 sum_att(f) 1 sum_rep(f) 2 n_rep(u) 3 coward(f) 4 beta_noise(f)
//  5 noise_cnt(u) 6 m_cnt(u) 7 e_sum(f) 8 p_sum(f) 9 pid_sum(f)

#define ZERO_U32 (16 + K_OBJ*2 + K_OBJ)   // accum + keys(as 2 u32) + attcnt

// ---- wave32 XOR-butterfly helpers via ds_swizzle_b32 (no LDS memory used) ---
// group-of-32 mode: offset = (xor_mask << 10) | (or=0 << 5) | (and=0x1F)
__device__ __forceinline__ float wave32_sum_f32(float v) {
    int b = __float_as_int(v);
    v += __int_as_float(__builtin_amdgcn_ds_swizzle(b, 0x401F)); b = __float_as_int(v); // xor 16
    v += __int_as_float(__builtin_amdgcn_ds_swizzle(b, 0x201F)); b = __float_as_int(v); // xor 8
    v += __int_as_float(__builtin_amdgcn_ds_swizzle(b, 0x101F)); b = __float_as_int(v); // xor 4
    v += __int_as_float(__builtin_amdgcn_ds_swizzle(b, 0x081F)); b = __float_as_int(v); // xor 2
    v += __int_as_float(__builtin_amdgcn_ds_swizzle(b, 0x041F));                        // xor 1
    return v;
}
__device__ __forceinline__ unsigned int wave32_sum_u32(unsigned int v) {
    v += (unsigned int)__builtin_amdgcn_ds_swizzle((int)v, 0x401F);
    v += (unsigned int)__builtin_amdgcn_ds_swizzle((int)v, 0x201F);
    v += (unsigned int)__builtin_amdgcn_ds_swizzle((int)v, 0x101F);
    v += (unsigned int)__builtin_amdgcn_ds_swizzle((int)v, 0x081F);
    v += (unsigned int)__builtin_amdgcn_ds_swizzle((int)v, 0x041F);
    return v;
}

__global__ void oc_init_kernel(unsigned char* ws) {
    int i = blockIdx.x * blockDim.x + threadIdx.x;
    if (i < ZERO_U32) ((unsigned int*)ws)[i] = 0u;
}

// -------------------- per-hit pass --------------------
__global__ void __launch_bounds__(256)
oc_perhit_kernel(const float* __restrict__ beta, const float* __restrict__ x,
                 const int* __restrict__ oid,
                 const float* __restrict__ energy, const float* __restrict__ eph,
                 const float* __restrict__ mom, const float* __restrict__ momph,
                 const float* __restrict__ logits, const int* __restrict__ pid,
                 unsigned char* __restrict__ ws)
{
    int i = blockIdx.x * blockDim.x + threadIdx.x;
    float* accf = (float*)(ws + OFF_ACC);
    unsigned int* accu = (unsigned int*)(ws + OFF_ACC);
    unsigned long long* keys = (unsigned long long*)(ws + OFF_KEYS);
    unsigned int* attcnt = (unsigned int*)(ws + OFF_ATTCNT);
    float* q   = (float*)(ws + OFF_Q);
    float* xn2 = (float*)(ws + OFF_XN2);

    float e_term = 0.f, p_term = 0.f, pid_term = 0.f, bnoise = 0.f;
    unsigned int mcnt = 0u, ncnt = 0u;

    if (i < N_HITS) {
        float b  = beta[i];
        float at = 0.5f * logf((1.0f + b) / (1.0f - b));   // arctanh
        float qi = at * at + 1.0f;                          // + Q_MIN
        float s2 = 0.f;
        #pragma unroll
        for (int d = 0; d < D_DIM; ++d) { float v = x[i*D_DIM + d]; s2 += v*v; }
        q[i]   = qi;
        xn2[i] = s2;

        int o = oid[i];
        if (o >= 1 && o <= K_OBJ) {
            unsigned int qb = __float_as_uint(qi);          // qi>=1 -> monotone bits
            unsigned long long key =
                ((unsigned long long)qb << 32) | (unsigned int)(~(unsigned int)i);
            atomicMax(&keys[o-1], key);
            atomicAdd(&attcnt[o-1], 1u);
        }
        if (o > 0) {  // non-noise: payload losses
            mcnt = 1u;
            float de = energy[i] - eph[i];
            e_term = de * de;
            #pragma unroll
            for (int d = 0; d < 3; ++d) {
                float dp = mom[i*3 + d] - momph[i*3 + d];
                p_term += dp * dp;
            }
            float l[C_CLS];
            float mx = -3.4e38f;
            #pragma unroll
            for (int c = 0; c < C_CLS; ++c) { l[c] = logits[i*C_CLS + c]; mx = fmaxf(mx, l[c]); }
            float se = 0.f;
            #pragma unroll
            for (int c = 0; c < C_CLS; ++c) se += expf(l[c] - mx);
            float lse = mx + logf(se);
            pid_term = lse - l[pid[i]];                     // NLL
        } else {
            ncnt = 1u;
            bnoise = b;
        }
    }

    // wave32 butterfly reductions (ds_swizzle), one atomic set per wave
    e_term   = wave32_sum_f32(e_term);
    p_term   = wave32_sum_f32(p_term);
    pid_term = wave32_sum_f32(pid_term);
    bnoise   = wave32_sum_f32(bnoise);
    mcnt     = wave32_sum_u32(mcnt);
    ncnt     = wave32_sum_u32(ncnt);
    if ((threadIdx.x & 31) == 0) {
        atomicAdd(&accf[7], e_term);
        atomicAdd(&accf[8], p_term);
        atomicAdd(&accf[9], pid_term);
        atomicAdd(&accf[4], bnoise);
        atomicAdd(&accu[6], mcnt);
        atomicAdd(&accu[5], ncnt);
    }
}

// -------------------- gather condensation points --------------------
__global__ void oc_gather_kernel(const float* __restrict__ x,
                                 const float* __restrict__ beta,
                                 unsigned char* __restrict__ ws)
{
    int k = blockIdx.x * blockDim.x + threadIdx.x;
    if (k >= K_OBJ) return;
    const unsigned long long* keys = (const unsigned long long*)(ws + OFF_KEYS);
    const unsigned int* attcnt = (const unsigned int*)(ws + OFF_ATTCNT);
    const float* q   = (const float*)(ws + OFF_Q);
    const float* xn2 = (const float*)(ws + OFF_XN2);
    float* qk      = (float*)(ws + OFF_QK);
    float* invAtt  = (float*)(ws + OFF_INVATT);
    float* invNrep = (float*)(ws + OFF_INVNREP);
    float* xkn2    = (float*)(ws + OFF_XKN2);
    float* xk      = (float*)(ws + OFF_XK);
    float* accf    = (float*)(ws + OFF_ACC);

    unsigned long long key = keys[k];
    unsigned int alpha = ~(unsigned int)(key & 0xFFFFFFFFull);  // first-max index
    qk[k]   = q[alpha];
    xkn2[k] = xn2[alpha];
    #pragma unroll
    for (int d = 0; d < D_DIM; ++d) xk[k*D_DIM + d] = x[alpha*D_DIM + d];
    float cnt = (float)attcnt[k];
    invAtt[k]  = 1.0f / (cnt + EPSF);
    invNrep[k] = 1.0f / (((float)N_HITS - cnt) + EPSF);
    atomicAdd(&accf[3], 1.0f - beta[alpha]);                    // coward sum
}

// -------------------- WMMA tile pass over the N x K matrix --------------------
// grid: 1024 blocks x 128 threads (4 waves). blockIdx%32 selects the 16-col
// tile; (blockIdx/32, waveId) -> 128 waves strided over the 7500 row tiles.
__global__ void __launch_bounds__(128)
oc_tile_kernel(const float* __restrict__ x, const int* __restrict__ oid,
               unsigned char* __restrict__ ws)
{
    const float* q      = (const float*)(ws + OFF_Q);
    const float* xn2    = (const float*)(ws + OFF_XN2);
    const float* xk     = (const float*)(ws + OFF_XK);
    const float* qk     = (const float*)(ws + OFF_QK);
    const float* invAtt = (const float*)(ws + OFF_INVATT);
    const float* invNrep= (const float*)(ws + OFF_INVNREP);
    const float* xkn2   = (const float*)(ws + OFF_XKN2);
    float* accf = (float*)(ws + OFF_ACC);
    unsigned int* accu = (unsigned int*)(ws + OFF_ACC);

    const int lane   = threadIdx.x & 31;
    const int waveId = threadIdx.x >> 5;
    const int half   = lane >> 4;     // 0|1: selects K-pair (A) / M-group (C)
    const int lp     = lane & 15;     // A: row M, B: col N

    const int colTile    = blockIdx.x & 31;                 // 32 column tiles
    const int waveLinear = (blockIdx.x >> 5) * 4 + waveId;  // 0..127 per column
    const int kcol  = colTile * 16 + lp;                    // this lane's column
    const int colId = kcol + 1;                             // object id 1..K

    // loop-invariant B fragments (x_k^T) + column scalars
    // B 4x16 layout: lane -> N=lp, vgpr v -> K = 2*half + v
    v2f b0 = *(const v2f*)(xk + kcol*D_DIM + 2*half);       // K cols 0..3
    v2f b1 = *(const v2f*)(xk + kcol*D_DIM + 4 + 2*half);   // K cols 4..7
    const float qk_l   = qk[kcol];
    const float xkn2_l = xkn2[kcol];
    const float invA_l = invAtt[kcol];
    const float invR_l = invNrep[kcol];

    float satt = 0.f, srep = 0.f;
    unsigned int crep = 0u;

    const int ROWTILES = N_HITS / 16;   // 7500, exact
    for (int rt = waveLinear; rt < ROWTILES; rt += 128) {
        const int i0 = rt * 16;
        // A 16x4 layout: lane -> M=lp, vgpr v -> K = 2*half + v
        const int arow = i0 + lp;
        v2f a0 = *(const v2f*)(x + arow*D_DIM + 2*half);
        v2f a1 = *(const v2f*)(x + arow*D_DIM + 4 + 2*half);

        v8f acc = {};
        acc = __builtin_amdgcn_wmma_f32_16x16x4_f32(
                  false, a0, false, b0, (short)0, acc, false, false);
        acc = __builtin_amdgcn_wmma_f32_16x16x4_f32(
                  false, a1, false, b1, (short)0, acc, false, false);

        // row scalars: this lane's 8 rows are i0 + 8*half + 0..7 (contiguous)
        const int ri = i0 + 8*half;
        float4 qA = *(const float4*)(q + ri);
        float4 qB = *(const float4*)(q + ri + 4);
        float4 nA = *(const float4*)(xn2 + ri);
        float4 nB = *(const float4*)(xn2 + ri + 4);
        int4   oA = *(const int4*)(oid + ri);
        int4   oB = *(const int4*)(oid + ri + 4);
        float qv[8] = {qA.x,qA.y,qA.z,qA.w, qB.x,qB.y,qB.z,qB.w};
        float nv[8] = {nA.x,nA.y,nA.z,nA.w, nB.x,nB.y,nB.z,nB.w};
        int   ov[8] = {oA.x,oA.y,oA.z,oA.w, oB.x,oB.y,oB.z,oB.w};

        // WGP-scope prefetch of next row tile: rows are re-read by the other
        // 31 column tiles, so pull into all cache levels (locality 3)
        int nrt = rt + 128;
        if (nrt < ROWTILES) {
            __builtin_prefetch(x + (nrt*16 + lp)*D_DIM, 0, 3);
            __builtin_prefetch(q + nrt*16 + 8*half, 0, 3);
        }

        #pragma unroll
        for (int j = 0; j < 8; ++j) {   // C/D: vgpr j -> M = j + 8*half, N = lp
            float dot  = acc[j];
            float d2   = nv[j] + xkn2_l - 2.0f * dot;
            d2         = fmaxf(d2, 0.0f);
            float dist = sqrtf(fmaxf(d2, 1e-12f));
            float qw   = qv[j] * qk_l;
            bool att   = (ov[j] == colId);
            satt += att ? qw * d2 * invA_l : 0.0f;
            bool rep   = (!att) && (dist < 1.0f);
            srep += rep ? qw * (1.0f - dist) * invR_l : 0.0f;
            crep += rep ? 1u : 0u;
        }
    }

    // wave32 butterfly reduction, one atomic triple per wave
    satt = wave32_sum_f32(satt);
    srep = wave32_sum_f32(srep);
    crep = wave32_sum_u32(crep);
    if (lane == 0) {
        atomicAdd(&accf[0], satt);
        atomicAdd(&accf[1], srep);
        atomicAdd(&accu[2], crep);
    }
}

// -------------------- finalize 10 scalars --------------------
__global__ void oc_finalize_kernel(const unsigned char* __restrict__ ws,
                                   float* __restrict__ out)
{
    const float* accf = (const float*)(ws + OFF_ACC);
    const unsigned int* accu = (const unsigned int*)(ws + OFF_ACC);
    float Kf = (float)K_OBJ;
    float v_att    = accf[0] / Kf;
    float v_rep    = accf[1] / Kf;
    float l_coward = accf[3] / Kf;
    float l_noise  = accf[4] / (float)accu[5];
    float n_rep    = (float)accu[2];
    float oc_total = v_att + v_rep + l_coward + l_noise;
    float nm       = (float)accu[6];
    float e_loss   = accf[7] / nm;
    float p_loss   = accf[8] / (nm * 3.0f);
    float pid_loss = accf[9] / nm;
    float total    = oc_total + (e_loss + p_loss + pid_loss);
    out[0] = v_att;    out[1] = v_rep;   out[2] = l_coward; out[3] = l_noise;
    out[4] = n_rep;    out[5] = oc_total;out[6] = e_loss;   out[7] = p_loss;
    out[8] = pid_loss; out[9] = total;
}

// ---------------------------------------------------------------------------
extern "C" void kernel_launch(void* const* d_in, const int* in_sizes, int n_in,
                              void* d_out, int out_size, void* d_ws, size_t ws_size,
                              hipStream_t stream)
{
    (void)in_sizes; (void)n_in; (void)out_size; (void)ws_size;
    const float* beta   = (const float*)d_in[0];
    const float* x      = (const float*)d_in[1];
    const int*   oid    = (const int*)d_in[2];
    const float* energy = (const float*)d_in[3];
    const float* eph    = (const float*)d_in[4];
    const float* mom    = (const float*)d_in[5];
    const float* momph  = (const float*)d_in[6];
    const float* logits = (const float*)d_in[7];
    const int*   pid    = (const int*)d_in[8];
    unsigned char* ws   = (unsigned char*)d_ws;
    float* out          = (float*)d_out;

    oc_init_kernel<<<(ZERO_U32 + 255) / 256, 256, 0, stream>>>(ws);
    oc_perhit_kernel<<<(N_HITS + 255) / 256, 256, 0, stream>>>(
        beta, x, oid, energy, eph, mom, momph, logits, pid, ws);
    oc_gather_kernel<<<(K_OBJ + 255) / 256, 256, 0, stream>>>(x, beta, ws);
    oc_tile_kernel<<<1024, 128, 0, stream>>>(x, oid, ws);
    oc_finalize_kernel<<<1, 1, 0, stream>>>(ws, out);
}